// ChamferLoss_6433861009633
// MI455X (gfx1250) — compile-verified
//
#include <hip/hip_runtime.h>
#include <hip/hip_bf16.h>
#include <float.h>

// CDNA5 / gfx1250, wave32.
typedef __attribute__((ext_vector_type(2))) float v2f;
typedef __attribute__((ext_vector_type(8))) float v8f;

#define NPTS         8192
#define DBCHUNK      2048                 // db points staged in LDS per chunk
#define PAD2         16                   // 16 float2 = 128 B pad -> +32 bank shift
#define WAVES_PER_WG 8
#define QPW          32                   // queries per wave: 2 WMMA N-tiles
#define QPB          (QPW * WAVES_PER_WG) // 256 queries per workgroup

// Direct v_min3_num_f32: avoids the compiler's canonicalize (v_max_num x,x)
// insertions when folding fminf chains into min3.
__device__ __forceinline__ float min3f(float a, float b, float c) {
    float d;
    asm("v_min3_num_f32 %0, %1, %2, %3" : "=v"(d) : "v"(a), "v"(b), "v"(c));
    return d;
}

// For each query q, find min over db points s of  ||s||^2 - 2<s,q>
// (the ||q||^2 constant is added after the min). WMMA mapping (16x16x4 f32):
//   A[m, 0..3] = [x0, x1, x2, ||s_m||^2]   <- ||s||^2 precomputed at LDS staging
//   B[0..3, n] = [-2q0, -2q1, -2q2, 1]
//   D[m,n]     = ||s_m||^2 - 2<s_m, q_n>
// A-frag layout (32-bit A 16x4): lanes 0-15 carry K0,K1 ; lanes 16-31 carry
// K2,K3 ; M = lane%16.  LDS keeps the two K-halves as separate float2 arrays
// (lo = (x0,x1), hi = (x2,rs)), so each lane's ds_load_b64 IS its A fragment:
// no per-iteration selects, and lo/hi arrays sit on disjoint bank halves.
// C/D layout: lane holds N=lane%16, VGPR v holds M = v + (lane>=16 ? 8 : 0)
// -> min over db = 4x v_min3_num_f32 per tile + one final shfl_xor(16).
__global__ __launch_bounds__(256)
void chamfer_min_kernel(const float* __restrict__ preds,
                        const float* __restrict__ gts,
                        float* __restrict__ ws) {
    // [0 .. DBCHUNK)                : lo half (x0,x1)   -> banks 0..31
    // [DBCHUNK+PAD2 .. 2*DBCHUNK+..): hi half (x2,rs)   -> banks 32..63
    __shared__ float2 sdb[2 * DBCHUNK + PAD2];

    const int lane = threadIdx.x & 31;
    const int wave = threadIdx.x >> 5;
    const int n    = lane & 15;          // column index within a 16-wide tile
    const bool hi  = lane >= 16;         // lanes 16..31 supply K=2,3
    const int b    = blockIdx.y;
    const int z    = blockIdx.z;         // 0: q=preds,db=gts ; 1: q=gts,db=preds

    const float* Q = z ? gts   : preds;
    const float* S = z ? preds : gts;
    float* out = ws + ((size_t)z * gridDim.y + b) * NPTS;

    const int qbase = blockIdx.x * QPB + wave * QPW;
    const int abase = (hi ? (DBCHUNK + PAD2) : 0) + n;   // this lane's A slice

    // ---- Build two B fragments (query tiles qbase+0..15 and qbase+16..31) ----
    const float* qp0 = Q + ((size_t)b * NPTS + (qbase + n)) * 3;
    const float* qp1 = qp0 + 16 * 3;
    const float a0 = qp0[0], a1 = qp0[1], a2 = qp0[2];
    const float b0 = qp1[0], b1 = qp1[1], b2 = qp1[2];
    const float rq0 = a0 * a0 + a1 * a1 + a2 * a2;
    const float rq1 = b0 * b0 + b1 * b1 + b2 * b2;
    v2f Bf0, Bf1;
    Bf0.x = hi ? (-2.0f * a2) : (-2.0f * a0);
    Bf0.y = hi ? 1.0f         : (-2.0f * a1);
    Bf1.x = hi ? (-2.0f * b2) : (-2.0f * b0);
    Bf1.y = hi ? 1.0f         : (-2.0f * b1);

    const float* sbase = S + (size_t)b * NPTS * 3;
    float runmin0 = FLT_MAX, runmin1 = FLT_MAX;

    for (int chunk = 0; chunk < NPTS; chunk += DBCHUNK) {
        // ---- Stage db chunk into LDS, split by K-half, rs precomputed ----
        #pragma unroll
        for (int i = threadIdx.x; i < DBCHUNK; i += 256) {
            const float* sp = sbase + (size_t)(chunk + i) * 3;
            float x0 = sp[0], x1 = sp[1], x2 = sp[2];
            float2 lo; lo.x = x0; lo.y = x1;
            float2 hv; hv.x = x2; hv.y = x0 * x0 + x1 * x1 + x2 * x2;
            sdb[i]                  = lo;
            sdb[DBCHUNK + PAD2 + i] = hv;
        }
        __syncthreads();

        // ---- 16 db points per step, two 16x16 WMMA tiles per step ----
        #pragma unroll 2
        for (int s = 0; s < DBCHUNK; s += 16) {
            float2 af = sdb[abase + s];            // ds_load_b64 == A fragment
            v2f Af; Af.x = af.x; Af.y = af.y;

            v8f c0 = {};
            c0 = __builtin_amdgcn_wmma_f32_16x16x4_f32(
                    false, Af, false, Bf0, (short)0, c0, false, false);
            v8f c1 = {};
            c1 = __builtin_amdgcn_wmma_f32_16x16x4_f32(
                    false, Af, false, Bf1, (short)0, c1, false, false);

            // 8 values + carry per tile -> exactly 4x v_min3_num_f32.
            runmin0 = min3f(min3f(c0[0], c0[1], c0[2]),
                            min3f(c0[3], c0[4], c0[5]),
                            min3f(c0[6], c0[7], runmin0));
            runmin1 = min3f(min3f(c1[0], c1[1], c1[2]),
                            min3f(c1[3], c1[4], c1[5]),
                            min3f(c1[6], c1[7], runmin1));
        }
        __syncthreads();
    }

    // Lane L and L+16 hold the same query column, disjoint M-halves: combine.
    runmin0 = fminf(runmin0, __shfl_xor(runmin0, 16, 32));
    runmin1 = fminf(runmin1, __shfl_xor(runmin1, 16, 32));

    if (!hi) {
        out[qbase + n]      = runmin0 + rq0;
        out[qbase + 16 + n] = runmin1 + rq1;
    }
}

// Deterministic single-block tree reduction: sum of all per-point mins.
__global__ __launch_bounds__(256)
void chamfer_reduce_kernel(const float* __restrict__ mins, int count,
                           float* __restrict__ out) {
    __shared__ float smem[256];
    float acc = 0.0f;
    for (int i = threadIdx.x; i < count; i += 256)
        acc += mins[i];
    smem[threadIdx.x] = acc;
    __syncthreads();
    for (int off = 128; off > 0; off >>= 1) {
        if ((int)threadIdx.x < off)
            smem[threadIdx.x] += smem[threadIdx.x + off];
        __syncthreads();
    }
    if (threadIdx.x == 0)
        out[0] = smem[0];
}

extern "C" void kernel_launch(void* const* d_in, const int* in_sizes, int n_in,
                              void* d_out, int out_size, void* d_ws, size_t ws_size,
                              hipStream_t stream) {
    const float* preds = (const float*)d_in[0];   // [B, N, 3]
    const float* gts   = (const float*)d_in[1];   // [B, N, 3]
    const int B = in_sizes[0] / (NPTS * 3);

    // ws layout: [B*NPTS] per-pred mins (z=0) | [B*NPTS] per-gt mins (z=1)
    float* mins = (float*)d_ws;

    dim3 grid(NPTS / QPB, B, 2);
    chamfer_min_kernel<<<grid, 256, 0, stream>>>(preds, gts, mins);

    chamfer_reduce_kernel<<<1, 256, 0, stream>>>(mins, 2 * B * NPTS, (float*)d_out);
}